// QuLinear_16930761081528
// MI455X (gfx1250) — compile-verified
//
#include <hip/hip_runtime.h>
#include <hip/hip_bf16.h>

// ---------------------------------------------------------------------------
// 16-qubit batched statevector sim on gfx1250, fused two-gate passes.
//
//  * YZY per qubit -> SU(2) 2x2; 4 bit-positions -> fused 16x16 complex gate.
//  * A layer = 2 passes (not 4): each pass applies TWO 16x16 gates to an
//    8-bit axis group via back-to-back 16x16x16 complex GEMMs
//    (v_wmma_f32_16x16x4_f32, K chained x4, complex = 4 real GEMMs),
//    with a per-wave LDS transpose between them.
//  * Pass A handles bits 0-7 (contiguous rows); pass B handles bits 8-15
//    (strided rows, staged through LDS so global traffic stays 128B-coalesced).
//  * CNOT ring = GF(2) suffix-XOR index permutation, fused into pass B stores.
//  * State (128 MB) is L2-resident (192 MB): 10 passes x 256 MB of L2 traffic
//    vs 21 passes for the unfused version.
// ---------------------------------------------------------------------------

typedef float v2f __attribute__((ext_vector_type(2)));
typedef float v8f __attribute__((ext_vector_type(8)));

#define W_MUL 0.6324555320336759f   // sqrt(2)/sqrt(5)
#define RS 258                      // tile row stride (float2), bank-spread
#define TS 18                       // transpose scratch row stride (float2)

// CNOT-ring permutation on the 16-bit amplitude index (bit p <-> qubit 15-p):
//   y_p = x_p ^ ... ^ x_15 (p<15),  y_15 = x_0 ^ ... ^ x_14
__device__ __forceinline__ unsigned perm16(unsigned x) {
    unsigned t = x ^ (x >> 1);
    t ^= t >> 2;  t ^= t >> 4;  t ^= t >> 8;
    return (t & 0x7FFFu) | (((t ^ (x >> 15)) & 1u) << 15);
}

// XOR-swizzled tile offset: element r of row rho; spreads the strided
// 16-element fragment reads across distinct LDS banks.
__device__ __forceinline__ unsigned ldsIdx(unsigned rho, unsigned r) {
    return rho * RS + (r ^ ((r >> 4) & 15u));
}

// ---------------------------------------------------------------------------
// Build the 20 fused 16x16 complex gates (5 layers x 4 bit-groups).
// Per (layer,group): 256 floats Gr row-major, then 256 floats Gi.
// Group g covers bit positions 4g..4g+3; local bit j <-> qubit 15-4g-j.
// ---------------------------------------------------------------------------
__global__ __launch_bounds__(256) void build_gates(const float* __restrict__ w,
                                                   float* __restrict__ gates) {
    const int lg = blockIdx.x;            // 0..19
    const int l  = lg >> 2, g = lg & 3;
    const int e  = threadIdx.x;           // 0..255
    const int mo = e >> 4, mi = e & 15;

    float pr = 1.0f, pi = 0.0f;
    #pragma unroll
    for (int j = 0; j < 4; ++j) {
        const int q = 15 - (4 * g + j);
        const float a = w[l * 48 + 3 * q + 0] * W_MUL;
        const float b = w[l * 48 + 3 * q + 1] * W_MUL;
        const float c = w[l * 48 + 3 * q + 2] * W_MUL;
        float sa, ca, sb, cb, sc, cc;
        __sincosf(0.5f * a, &sa, &ca);
        __sincosf(0.5f * b, &sb, &cb);
        __sincosf(0.5f * c, &sc, &cc);
        const int bo = (mo >> j) & 1, bi = (mi >> j) & 1;
        float ure, uim;                   // U = Ry(c) * Rz(b) * Ry(a)
        if (!bo && !bi)      { ure =  cb * (cc * ca - sc * sa); uim = -sb * (cc * ca + sc * sa); }
        else if (!bo &&  bi) { ure = -cb * (cc * sa + sc * ca); uim =  sb * (cc * sa - sc * ca); }
        else if ( bo && !bi) { ure =  cb * (sc * ca + cc * sa); uim =  sb * (cc * sa - sc * ca); }
        else                 { ure =  cb * (cc * ca - sc * sa); uim =  sb * (cc * ca + sc * sa); }
        const float nr = pr * ure - pi * uim;
        const float ni = pr * uim + pi * ure;
        pr = nr; pi = ni;
    }
    gates[lg * 512 + e]       = pr;
    gates[lg * 512 + 256 + e] = pi;
}

// ---------------------------------------------------------------------------
// Fused pass: apply gate gA (low 4 bits of an 8-bit axis group) then gB
// (high 4 bits) to every 256-amplitude row of that group.
//   passB=0: group = bits 0-7  (rows contiguous, block tile = 16 hi values)
//   passB=1: group = bits 8-15 (rows strided 256, block tile = 16 lo values)
// real_in: stage real f32 input as (x, 0).  do_perm: CNOT perm on stores.
// Block = 256 thr = 8 waves; wave does 2 rows; 32 complex-GEMM WMMAs per row.
// ---------------------------------------------------------------------------
__global__ __launch_bounds__(256) void fused_pass(const float2* src,
                                                  const float*  srcReal,
                                                  float2*       dst,
                                                  const float* __restrict__ gA,
                                                  const float* __restrict__ gB,
                                                  int passB, int do_perm, int real_in) {
    __shared__ float2 tile[16 * RS];        // 33,024 B staged state tile
    __shared__ float2 xpose[8 * 16 * TS];   // 18,432 B per-wave transpose scratch

    const unsigned t    = threadIdx.x;
    const unsigned lane = t & 31u;
    const unsigned wv   = t >> 5;
    const unsigned n    = lane & 15u;       // matrix N index
    const unsigned h    = lane >> 4;        // lane half
    const unsigned blk  = blockIdx.x;
    const unsigned b    = blk >> 4;         // batch element
    const unsigned sub  = blk & 15u;        // tile index within batch
    const unsigned boff = b << 16;

    // ---- stage global -> LDS (coalesced) ----
    if (passB) {
        const unsigned ll  = t & 15u;       // row (lo value) this thread feeds
        const unsigned rr0 = t >> 4;
        const unsigned base = boff | (sub << 4);
        #pragma unroll
        for (int j = 0; j < 16; ++j) {
            const unsigned r = (unsigned)j * 16u + rr0;
            tile[ldsIdx(ll, r)] = src[base + (r << 8) + ll];
        }
    } else {
        const unsigned base = boff | (sub << 12);
        if (real_in) {
            #pragma unroll
            for (int j = 0; j < 16; ++j) {
                float2 v; v.x = srcReal[base + (unsigned)j * 256u + t]; v.y = 0.0f;
                tile[ldsIdx((unsigned)j, t)] = v;
            }
        } else {
            #pragma unroll
            for (int j = 0; j < 16; ++j)
                tile[ldsIdx((unsigned)j, t)] = src[base + (unsigned)j * 256u + t];
        }
    }
    __syncthreads();

    // ---- A fragments for both gates: A[M=lane&15][K = 4kk + 2h + v] ----
    v2f a1r[4], a1i[4], a2r[4], a2i[4];
    #pragma unroll
    for (int kk = 0; kk < 4; ++kk) {
        const unsigned K0 = 4u * kk + 2u * h;
        a1r[kk].x = gA[n * 16u + K0];        a1r[kk].y = gA[n * 16u + K0 + 1u];
        a1i[kk].x = gA[256u + n * 16u + K0]; a1i[kk].y = gA[256u + n * 16u + K0 + 1u];
        a2r[kk].x = gB[n * 16u + K0];        a2r[kk].y = gB[n * 16u + K0 + 1u];
        a2i[kk].x = gB[256u + n * 16u + K0]; a2i[kk].y = gB[256u + n * 16u + K0 + 1u];
    }

    float2* xp = &xpose[wv * (16u * TS)];

    #pragma unroll
    for (int rr = 0; rr < 2; ++rr) {
        const unsigned rho = wv * 2u + (unsigned)rr;   // row within tile

        // B1 fragments: B[K=m0][N=c] = row[c*16 + m0]
        v2f br[4], bi[4];
        #pragma unroll
        for (int kk = 0; kk < 4; ++kk) {
            const unsigned K0 = 4u * kk + 2u * h;
            const float2 f0 = tile[ldsIdx(rho, n * 16u + K0)];
            const float2 f1 = tile[ldsIdx(rho, n * 16u + K0 + 1u)];
            br[kk].x = f0.x; br[kk].y = f1.x;
            bi[kk].x = f0.y; bi[kk].y = f1.y;
        }

        // GEMM1: out0 = gA x row  (complex, K=16)
        v8f dr = {0.f,0.f,0.f,0.f,0.f,0.f,0.f,0.f};
        v8f di = {0.f,0.f,0.f,0.f,0.f,0.f,0.f,0.f};
        #pragma unroll
        for (int kk = 0; kk < 4; ++kk)
            dr = __builtin_amdgcn_wmma_f32_16x16x4_f32(false, a1r[kk], false, br[kk], (short)0, dr, false, false);
        #pragma unroll
        for (int kk = 0; kk < 4; ++kk) {
            v2f an; an.x = -a1i[kk].x; an.y = -a1i[kk].y;
            dr = __builtin_amdgcn_wmma_f32_16x16x4_f32(false, an, false, bi[kk], (short)0, dr, false, false);
        }
        #pragma unroll
        for (int kk = 0; kk < 4; ++kk)
            di = __builtin_amdgcn_wmma_f32_16x16x4_f32(false, a1i[kk], false, br[kk], (short)0, di, false, false);
        #pragma unroll
        for (int kk = 0; kk < 4; ++kk)
            di = __builtin_amdgcn_wmma_f32_16x16x4_f32(false, a1r[kk], false, bi[kk], (short)0, di, false, false);

        // transpose out0 through per-wave LDS scratch: xp[M][N] = out0(M,N)
        #pragma unroll
        for (int p = 0; p < 8; ++p) {
            const unsigned M = (unsigned)p + 8u * h;
            float2 o; o.x = dr[p]; o.y = di[p];
            xp[M * TS + n] = o;
        }
        asm volatile("s_wait_dscnt 0x0" ::: "memory");

        // B2 fragments: B[K=m1][N=m0'] = out0[m0'][m1] = xp[N][K]
        v2f br2[4], bi2[4];
        #pragma unroll
        for (int kk = 0; kk < 4; ++kk) {
            const unsigned K0 = 4u * kk + 2u * h;
            const float2 g0 = xp[n * TS + K0];
            const float2 g1 = xp[n * TS + K0 + 1u];
            br2[kk].x = g0.x; br2[kk].y = g1.x;
            bi2[kk].x = g0.y; bi2[kk].y = g1.y;
        }

        // GEMM2: out1 = gB x out0^T
        v8f er = {0.f,0.f,0.f,0.f,0.f,0.f,0.f,0.f};
        v8f ei = {0.f,0.f,0.f,0.f,0.f,0.f,0.f,0.f};
        #pragma unroll
        for (int kk = 0; kk < 4; ++kk)
            er = __builtin_amdgcn_wmma_f32_16x16x4_f32(false, a2r[kk], false, br2[kk], (short)0, er, false, false);
        #pragma unroll
        for (int kk = 0; kk < 4; ++kk) {
            v2f an; an.x = -a2i[kk].x; an.y = -a2i[kk].y;
            er = __builtin_amdgcn_wmma_f32_16x16x4_f32(false, an, false, bi2[kk], (short)0, er, false, false);
        }
        #pragma unroll
        for (int kk = 0; kk < 4; ++kk)
            ei = __builtin_amdgcn_wmma_f32_16x16x4_f32(false, a2i[kk], false, br2[kk], (short)0, ei, false, false);
        #pragma unroll
        for (int kk = 0; kk < 4; ++kk)
            ei = __builtin_amdgcn_wmma_f32_16x16x4_f32(false, a2r[kk], false, bi2[kk], (short)0, ei, false, false);

        // store: element r' = M*16 + N of this row (bits0-3 = m0', bits4-7 = m1')
        #pragma unroll
        for (int p = 0; p < 8; ++p) {
            const unsigned M  = (unsigned)p + 8u * h;
            const unsigned rp = M * 16u + n;
            unsigned i = passB ? ((rp << 8) | (sub << 4) | rho)
                               : ((sub << 12) | (rho << 8) | rp);
            if (do_perm) i = perm16(i);
            float2 o; o.x = er[p]; o.y = ei[p];
            dst[boff + i] = o;
        }
    }
}

// ---------------------------------------------------------------------------
extern "C" void kernel_launch(void* const* d_in, const int* in_sizes, int n_in,
                              void* d_out, int out_size, void* d_ws, size_t ws_size,
                              hipStream_t stream) {
    const float* x = (const float*)d_in[0];      // [256,1,65536] f32
    const float* w = (const float*)d_in[1];      // [240] f32

    float*  gates = (float*)d_ws;                               // 40 KB
    float2* W     = (float2*)((char*)d_ws + (size_t)(1 << 16)); // 128 MB ping buffer
    float2* O     = (float2*)d_out;                             // complex64 interleaved

    build_gates<<<20, 256, 0, stream>>>(w, gates);

    const dim3 grid(4096), block(256);   // 4096 amplitudes per block
    #define GM(l, g) (gates + ((l) * 4 + (g)) * 512)

    // Layer 0: real input fused into pass A; CNOT perm fused into pass B store
    fused_pass<<<grid, block, 0, stream>>>(nullptr, x, W, GM(0, 0), GM(0, 1), 0, 0, 1);
    fused_pass<<<grid, block, 0, stream>>>(W, nullptr, O, GM(0, 2), GM(0, 3), 1, 1, 0);
    // Layer 1
    fused_pass<<<grid, block, 0, stream>>>(O, nullptr, O, GM(1, 0), GM(1, 1), 0, 0, 0);
    fused_pass<<<grid, block, 0, stream>>>(O, nullptr, W, GM(1, 2), GM(1, 3), 1, 1, 0);
    // Layer 2
    fused_pass<<<grid, block, 0, stream>>>(W, nullptr, W, GM(2, 0), GM(2, 1), 0, 0, 0);
    fused_pass<<<grid, block, 0, stream>>>(W, nullptr, O, GM(2, 2), GM(2, 3), 1, 1, 0);
    // Layer 3
    fused_pass<<<grid, block, 0, stream>>>(O, nullptr, O, GM(3, 0), GM(3, 1), 0, 0, 0);
    fused_pass<<<grid, block, 0, stream>>>(O, nullptr, W, GM(3, 2), GM(3, 3), 1, 1, 0);
    // Layer 4: final YZY, no CNOT ring -> d_out
    fused_pass<<<grid, block, 0, stream>>>(W, nullptr, W, GM(4, 0), GM(4, 1), 0, 0, 0);
    fused_pass<<<grid, block, 0, stream>>>(W, nullptr, O, GM(4, 2), GM(4, 3), 1, 0, 0);
    #undef GM
}